// NFPLayer_49916109914405
// MI455X (gfx1250) — compile-verified
//
#include <hip/hip_runtime.h>
#include <math.h>

// Problem constants (from reference)
#define A_N    4096
#define NA_N   62
#define NB_N   6
#define R_N    4
#define OUT_N  2048
#define HLEN_N 68   // NA + NB

typedef __attribute__((ext_vector_type(2))) float v2f;
typedef __attribute__((ext_vector_type(8))) float v8f;

// fp32 WMMA: D(16x16) = A(16x4) * B(4x16) + C    -> v_wmma_f32_16x16x4_f32
__device__ __forceinline__ v8f wmma_f32(v2f a, v2f b, v8f c) {
  return __builtin_amdgcn_wmma_f32_16x16x4_f32(
      /*neg_a=*/false, a, /*neg_b=*/false, b,
      /*c_mod=*/(short)0, c, /*reuse_a=*/false, /*reuse_b=*/false);
}

// ---------------------------------------------------------------------------
// bond_[i,b] = sum_j adj[i,j] * bond_feature[j,i,b]
// Exploits sparsity: each thread scans 16 contiguous adj entries of row i and
// gathers bond_feature only for nonzeros (~1.6 MB instead of 402 MB).
// Fixed-order LDS tree reduction -> deterministic.
// ---------------------------------------------------------------------------
__global__ __launch_bounds__(256) void bond_kernel(const float* __restrict__ adj,
                                                   const float* __restrict__ bf,
                                                   float* __restrict__ bond) {
  __shared__ float red[256 * NB_N];
  const int i = blockIdx.x;
  const int tid = threadIdx.x;
  float acc[NB_N];
#pragma unroll
  for (int b = 0; b < NB_N; ++b) acc[b] = 0.0f;
  const float* arow = adj + (size_t)i * A_N;
  const int j0 = tid * 16;
  for (int jj = 0; jj < 16; ++jj) {
    float w = arow[j0 + jj];
    if (w != 0.0f) {
      const float* src = bf + ((size_t)(j0 + jj) * A_N + i) * NB_N;
#pragma unroll
      for (int b = 0; b < NB_N; ++b) acc[b] += w * src[b];
    }
  }
#pragma unroll
  for (int b = 0; b < NB_N; ++b) red[tid * NB_N + b] = acc[b];
  __syncthreads();
  for (int s = 128; s > 0; s >>= 1) {
    if (tid < s) {
#pragma unroll
      for (int b = 0; b < NB_N; ++b)
        red[tid * NB_N + b] += red[(tid + s) * NB_N + b];
    }
    __syncthreads();
  }
  if (tid < NB_N) bond[i * NB_N + tid] = red[tid];
}

// ---------------------------------------------------------------------------
// v[i, 0:6]  = bond_[i]
// v[i, 6:68] = atom_[i,:] = sum_{j: adj[i,j]!=0} adj[i,j] * fprev[j, 0:62]
// adj row staged in LDS (L2-resident after first round). 4 j-quarters x 64
// column-threads; deterministic per-thread accumulation + fixed-order reduce.
// ---------------------------------------------------------------------------
__global__ __launch_bounds__(256) void gather_kernel(const float* __restrict__ adj,
                                                     const float* __restrict__ fprev,
                                                     int strideF,
                                                     const float* __restrict__ bond,
                                                     float* __restrict__ vbuf) {
  __shared__ alignas(16) float adjrow[A_N];
  __shared__ float colred[4 * 64];
  const int i = blockIdx.x;
  const int tid = threadIdx.x;
  const float4* arow4 = (const float4*)(adj + (size_t)i * A_N);
  for (int idx = tid; idx < A_N / 4; idx += 256)
    ((float4*)adjrow)[idx] = arow4[idx];
  __syncthreads();
  const int c = tid & 63;   // output column
  const int g = tid >> 6;   // j-range quarter
  float acc = 0.0f;
  const int jb = g * (A_N / 4);
  if (c < NA_N) {
    for (int jj = 0; jj < A_N / 4; ++jj) {
      float w = adjrow[jb + jj];           // LDS broadcast; branch is uniform
      if (w != 0.0f) acc += w * fprev[(size_t)(jb + jj) * strideF + c];
    }
  }
  colred[g * 64 + c] = acc;
  __syncthreads();
  if (tid < 64) {
    float s = colred[tid] + colred[64 + tid] + colred[128 + tid] + colred[192 + tid];
    if (tid < NA_N) vbuf[i * HLEN_N + NB_N + tid] = s;
    if (tid < NB_N) vbuf[i * HLEN_N + tid] = bond[i * NB_N + tid];
  }
}

// ---------------------------------------------------------------------------
// f = sigmoid(v @ H[r]) ; v: [4096,68], H: [68,68] -> f: [4096,68]
// One 16-row tile per block, 5 waves = 5 N-tiles (N padded to 80 with zeros).
// K = 68 = 17 steps of v_wmma_f32_16x16x4_f32.
// ---------------------------------------------------------------------------
__global__ __launch_bounds__(160) void h_kernel(const float* __restrict__ v,
                                                const float* __restrict__ Hm,
                                                float* __restrict__ f) {
  __shared__ float vt[16 * HLEN_N];     // 16 x 68
  __shared__ float Hs[HLEN_N * 80];     // 68 x 80 (cols 68..79 zero)
  const int tid = threadIdx.x;
  const int m0 = blockIdx.x * 16;
  for (int idx = tid; idx < 16 * HLEN_N; idx += 160)
    vt[idx] = v[(m0 + idx / HLEN_N) * HLEN_N + (idx % HLEN_N)];
  for (int idx = tid; idx < HLEN_N * 80; idx += 160) {
    int rr = idx / 80, cc = idx % 80;
    Hs[idx] = (cc < HLEN_N) ? Hm[rr * HLEN_N + cc] : 0.0f;
  }
  __syncthreads();
  const int wave = tid >> 5;
  const int lane = tid & 31;
  const int lo = lane & 15;
  const int hi = lane >> 4;
  const int n0 = wave * 16;
  v8f acc = {};
  for (int k0 = 0; k0 < HLEN_N; k0 += 4) {
    v2f a, b;
    a.x = vt[lo * HLEN_N + k0 + 2 * hi];
    a.y = vt[lo * HLEN_N + k0 + 2 * hi + 1];
    b.x = Hs[(k0 + 2 * hi) * 80 + n0 + lo];
    b.y = Hs[(k0 + 2 * hi + 1) * 80 + n0 + lo];
    acc = wmma_f32(a, b, acc);
  }
  const int col = n0 + lo;
  if (col < HLEN_N) {
#pragma unroll
    for (int p = 0; p < 8; ++p) {
      int row = m0 + p + hi * 8;
      f[row * HLEN_N + col] = 1.0f / (1.0f + __expf(-acc[p]));
    }
  }
}

// ---------------------------------------------------------------------------
// y = f @ W[r] (4096x2048); per-row softmax; column-sum over rows.
// 64 rows per block, 8 waves: wave = (nt-half, mtile). A-fragments for the
// whole K=68 are cached in registers (17 x v2f); W tiles stream from L2.
// |y| <= ~1 (sigmoid x glorot) => softmax without max-subtraction is safe,
// so pass 1 computes row sums of exp(y) and pass 2 recomputes y (cheap) and
// accumulates normalized column sums. All reductions fixed-order shuffles /
// single-writer LDS -> deterministic; block partials go to workspace.
// ---------------------------------------------------------------------------
__global__ __launch_bounds__(256) void w_kernel(const float* __restrict__ f,
                                                const float* __restrict__ Wm,
                                                float* __restrict__ colWS) {
  __shared__ float ft[64 * HLEN_N];     // 17.4 KB
  __shared__ float rspart[2][64];
  __shared__ float rsinv[64];
  __shared__ float colPart[4][OUT_N];   // 32 KB
  const int tid = threadIdx.x;
  const int mbase = blockIdx.x * 64;
  for (int idx = tid; idx < 64 * HLEN_N; idx += 256)
    ft[idx] = f[(mbase + idx / HLEN_N) * HLEN_N + (idx % HLEN_N)];
  __syncthreads();
  const int wave = tid >> 5, lane = tid & 31, lo = lane & 15, hi = lane >> 4;
  const int mtile = wave & 3;
  const int m0l = mtile * 16;
  const int ntBase = (wave >> 2) * 64;   // waves 0-3: tiles 0..63; 4-7: 64..127
  v2f afr[17];
#pragma unroll
  for (int kk = 0; kk < 17; ++kk) {
    afr[kk].x = ft[(m0l + lo) * HLEN_N + kk * 4 + 2 * hi];
    afr[kk].y = ft[(m0l + lo) * HLEN_N + kk * 4 + 2 * hi + 1];
  }
  float rs[8];
#pragma unroll
  for (int p = 0; p < 8; ++p) rs[p] = 0.0f;

  // pass 1: row sums of exp(y)
  for (int t = 0; t < 64; ++t) {
    const int n0 = (ntBase + t) * 16;
    const float* bp = Wm + n0 + lo;
    if (t + 1 < 64)  // prefetch next W tile region (gfx1250 global_prefetch_b8)
      __builtin_prefetch(Wm + (ntBase + t + 1) * 16 + lo, 0, 1);
    v8f acc = {};
#pragma unroll
    for (int kk = 0; kk < 17; ++kk) {
      v2f b;
      b.x = bp[(size_t)(kk * 4 + 2 * hi) * OUT_N];
      b.y = bp[(size_t)(kk * 4 + 2 * hi + 1) * OUT_N];
      acc = wmma_f32(afr[kk], b, acc);
    }
#pragma unroll
    for (int p = 0; p < 8; ++p) rs[p] += __expf(acc[p]);
  }
#pragma unroll
  for (int p = 0; p < 8; ++p) {
    float vv = rs[p];
#pragma unroll
    for (int m = 1; m < 16; m <<= 1) vv += __shfl_xor(vv, m, 32);
    if (lo == 0) rspart[wave >> 2][m0l + p + 8 * hi] = vv;
  }
  __syncthreads();
  if (tid < 64) rsinv[tid] = 1.0f / (rspart[0][tid] + rspart[1][tid]);
  __syncthreads();

  // pass 2: recompute y, normalize, column-sum over the 16 rows of this mtile
  for (int t = 0; t < 64; ++t) {
    const int n0 = (ntBase + t) * 16;
    const float* bp = Wm + n0 + lo;
    v8f acc = {};
#pragma unroll
    for (int kk = 0; kk < 17; ++kk) {
      v2f b;
      b.x = bp[(size_t)(kk * 4 + 2 * hi) * OUT_N];
      b.y = bp[(size_t)(kk * 4 + 2 * hi + 1) * OUT_N];
      acc = wmma_f32(afr[kk], b, acc);
    }
    float cp = 0.0f;
#pragma unroll
    for (int p = 0; p < 8; ++p)
      cp += __expf(acc[p]) * rsinv[m0l + p + 8 * hi];
    cp += __shfl_xor(cp, 16, 32);          // fold hi=0/1 halves (same column)
    if (hi == 0) colPart[mtile][n0 + lo] = cp;  // single writer per column
  }
  __syncthreads();
  for (int c = tid; c < OUT_N; c += 256)
    colWS[(size_t)blockIdx.x * OUT_N + c] =
        colPart[0][c] + colPart[1][c] + colPart[2][c] + colPart[3][c];
}

__global__ void zero_kernel(float* __restrict__ out, int n) {
  int c = blockIdx.x * blockDim.x + threadIdx.x;
  if (c < n) out[c] = 0.0f;
}

// Deterministic cross-block reduction of the 64 column partials into out.
__global__ void reduce_kernel(const float* __restrict__ colWS,
                              float* __restrict__ out) {
  int c = blockIdx.x * blockDim.x + threadIdx.x;
  if (c < OUT_N) {
    float s = 0.0f;
    for (int b = 0; b < A_N / 64; ++b) s += colWS[(size_t)b * OUT_N + c];
    out[c] += s;
  }
}

extern "C" void kernel_launch(void* const* d_in, const int* in_sizes, int n_in,
                              void* d_out, int out_size, void* d_ws, size_t ws_size,
                              hipStream_t stream) {
  const float* feature = (const float*)d_in[0];   // [4096, 62]
  const float* adj     = (const float*)d_in[1];   // [4096, 4096]
  const float* bf      = (const float*)d_in[2];   // [4096, 4096, 6]
  const float* Hm      = (const float*)d_in[3];   // [4, 68, 68]
  const float* Wm      = (const float*)d_in[4];   // [4, 68, 2048]
  float* out = (float*)d_out;                     // [2048] (flattened [1,2048])
  float* ws  = (float*)d_ws;

  float* bond  = ws;                          // 4096*6
  float* vbuf  = bond + A_N * NB_N;           // 4096*68
  float* fbuf  = vbuf + A_N * HLEN_N;         // 4096*68
  float* colWS = fbuf + A_N * HLEN_N;         // 64*2048

  zero_kernel<<<(OUT_N + 255) / 256, 256, 0, stream>>>(out, OUT_N);
  bond_kernel<<<A_N, 256, 0, stream>>>(adj, bf, bond);
  for (int r = 0; r < R_N; ++r) {
    const float* fprev = (r == 0) ? feature : fbuf;
    const int strideF  = (r == 0) ? NA_N : HLEN_N;
    gather_kernel<<<A_N, 256, 0, stream>>>(adj, fprev, strideF, bond, vbuf);
    h_kernel<<<A_N / 16, 160, 0, stream>>>(vbuf, Hm + r * HLEN_N * HLEN_N, fbuf);
    w_kernel<<<A_N / 64, 256, 0, stream>>>(fbuf, Wm + (size_t)r * HLEN_N * OUT_N, colWS);
    reduce_kernel<<<(OUT_N + 255) / 256, 256, 0, stream>>>(colWS, out);
  }
}